// ReLuPCA_15796889714879
// MI455X (gfx1250) — compile-verified
//
#include <hip/hip_runtime.h>
#include <cstdint>
#include <cstddef>

// Problem constants (from reference): x is (32, 256, 56, 56) f32.
#define C_DIM 256
#define HW_DIM 3136            // 56*56
#define N_IMG 32
#define M_DIM 100352           // N_IMG * HW_DIM
#define NTILES (M_DIM / 16)    // 6272 column tiles of 16
#define GEMM_GRID 784

typedef __attribute__((ext_vector_type(2))) float v2f;
typedef __attribute__((ext_vector_type(8))) float v8f;

// Monotone float <-> int encoding so min/max can use integer atomics.
__device__ __forceinline__ int fenc(float f) {
  int b = __float_as_int(f);
  return b >= 0 ? b : -(b & 0x7fffffff);
}
__device__ __forceinline__ float fdec(int e) {
  return e >= 0 ? __int_as_float(e)
                : __int_as_float((int)((unsigned)(-e) | 0x80000000u));
}

// ---------------------------------------------------------------- init
__global__ void k_init(float* sums, int* gmin, int* gmax) {
  int t = threadIdx.x;
  sums[t] = 0.f;
  gmin[t] = 0x7fffffff;
  gmax[t] = (int)0x80000000;
}

// ------------------------------------------------- per-channel relu sums
__global__ __launch_bounds__(256) void k_chansum(const float* __restrict__ x,
                                                 float* __restrict__ sums) {
  __shared__ float red[256];
  int plane = blockIdx.x;            // n*C + c
  int c = plane & (C_DIM - 1);
  const float* p = x + (size_t)plane * HW_DIM;
  float s = 0.f;
  for (int i = threadIdx.x; i < HW_DIM; i += 256) s += fmaxf(p[i], 0.f);
  red[threadIdx.x] = s;
  __syncthreads();
  for (int d = 128; d > 0; d >>= 1) {
    if (threadIdx.x < d) red[threadIdx.x] += red[threadIdx.x + d];
    __syncthreads();
  }
  if (threadIdx.x == 0) atomicAdd(&sums[c], red[0]);
}

// ------------------------------------------- mn = sums/M ; tvec = u^T mn
__global__ void k_mn_tvec(const float* __restrict__ u,
                          const float* __restrict__ sums,
                          float* __restrict__ mn, float* __restrict__ tvec) {
  __shared__ float smn[C_DIM];
  int t = threadIdx.x;
  float m = sums[t] * (1.f / (float)M_DIM);
  smn[t] = m;
  mn[t] = m;
  __syncthreads();
  float s = 0.f;
  for (int c = 0; c < C_DIM; ++c) s += u[c * C_DIM + t] * smn[c];
  tvec[t] = s;
}

// ---------------------------------------------------------------- GEMM
// PHASE1: P = clamp(u^T relu(x) - tvec, +-cv), track per-row min/max.
// PHASE2: out = u * q + off, scattered back to NCHW.
// Block = 512 threads = 16 wave32; wave w owns row tile w (rows 16w..16w+15).
// A fragments (16x4 f32 WMMA A layout: M=lane%16, K=v+2*(lane/16)) for all
// 64 k-steps are register resident (128 VGPRs). B tiles (256x16) are staged
// into double-buffered LDS in an interleaved-pair layout so every B fragment
// (K=v+2*(lane/16), N=lane%16) is one aligned 8-byte LDS load (the compiler
// merges pairs into ds_load_2addr_b64). Phase 2 stages with
// GLOBAL_LOAD_ASYNC_TO_LDS_B32 (ASYNCcnt-tracked DMA, VGPR bypass).
template <bool PHASE1>
__global__ __launch_bounds__(512, 1)
void k_gemm(const float* __restrict__ src,     // x (phase1) or P (phase2)
            const float* __restrict__ u,
            const float* __restrict__ rowsub,  // tvec (phase1) or off (phase2)
            const float* __restrict__ cv,      // clampVal (phase1 only)
            float* __restrict__ dst,           // P (phase1) or out (phase2)
            int* __restrict__ gmin, int* __restrict__ gmax) {
  __shared__ float sB[2][4096];                // double-buffered 256x16 tiles
  __shared__ int smin[C_DIM], smax[C_DIM];

  const int tid  = threadIdx.x;
  const int w    = tid >> 5;                   // wave id == row tile
  const int lane = tid & 31;
  const int half = lane >> 4;
  const int n    = lane & 15;
  const int co   = (w << 4) | n;               // A-fragment row (M index)
  const size_t RS = PHASE1 ? (size_t)HW_DIM : (size_t)M_DIM;  // B row stride

  if (PHASE1 && tid < C_DIM) {
    smin[tid] = 0x7fffffff;
    smax[tid] = (int)0x80000000;
  }

  // Register-resident A fragments for the whole K=256 loop.
  v2f Areg[64];
#pragma unroll
  for (int s = 0; s < 64; ++s) {
    int k = (s << 2) + (half << 1);
    if (PHASE1) {                               // A = u^T : A[co][k] = u[k][co]
      v2f a;
      a.x = u[k * C_DIM + co];
      a.y = u[(k + 1) * C_DIM + co];
      Areg[s] = a;
    } else {                                    // A = u : A[co][k] = u[co][k]
      Areg[s] = *(const v2f*)(u + (size_t)co * C_DIM + k);
    }
  }

  // Hoisted per-row epilogue constants (C/D layout: row = 16w + i + 8*half).
  float rsub[8];
  float cvv[8];
#pragma unroll
  for (int i = 0; i < 8; ++i) {
    int coI = (w << 4) + i + (half << 3);
    rsub[i] = rowsub[coI];
    if (PHASE1) cvv[i] = cv[coI];
  }

  // ---- staging helpers -------------------------------------------------
  auto tileBase = [&](int tile) -> const float* {
    int m0 = tile << 4;
    if (PHASE1) {
      int img = m0 / HW_DIM;
      int hw0 = m0 - img * HW_DIM;
      return src + (size_t)img * C_DIM * HW_DIM + hw0;
    }
    return src + m0;
  };

  // Phase 1: vectorized global prefetch into registers (relu fused on store).
  float4 pre0, pre1;
  auto issueLoads = [&](const float* bsrc) {
    int g0 = tid, g1 = tid + 512;               // groups of 4 consecutive cols
    pre0 = *(const float4*)(bsrc + (size_t)(g0 >> 2) * RS + ((g0 & 3) << 2));
    pre1 = *(const float4*)(bsrc + (size_t)(g1 >> 2) * RS + ((g1 & 3) << 2));
  };
  auto storeLds = [&](int b) {
    auto st = [&](int g, float4 v) {
      int c = g >> 2, j0 = (g & 3) << 2;
      float* d = &sB[b][((c >> 1) << 5) + (c & 1) + (j0 << 1)];
      d[0] = fmaxf(v.x, 0.f);
      d[2] = fmaxf(v.y, 0.f);
      d[4] = fmaxf(v.z, 0.f);
      d[6] = fmaxf(v.w, 0.f);
    };
    st(tid, pre0);
    st(tid + 512, pre1);
  };

  // Phase 2: async DMA global -> LDS (ASYNCcnt).
  auto asyncStage = [&](int b, const float* bsrc) {
#if defined(__gfx1250__)
#pragma unroll
    for (int e = tid; e < 4096; e += 512) {
      int c = e >> 4, j = e & 15;
      const float* g = bsrc + (size_t)c * RS + j;
      unsigned lo =
          (unsigned)(uintptr_t)&sB[b][((c >> 1) << 5) + (j << 1) + (c & 1)];
      asm volatile("global_load_async_to_lds_b32 %0, %1, off"
                   :
                   : "v"(lo), "v"(g)
                   : "memory");
    }
#else
    for (int e = tid; e < 4096; e += 512) {
      int c = e >> 4, j = e & 15;
      sB[b][((c >> 1) << 5) + (j << 1) + (c & 1)] = bsrc[(size_t)c * RS + j];
    }
#endif
  };
  auto asyncWait = [&]() {
#if defined(__gfx1250__)
    asm volatile("s_wait_asynccnt 0x0" ::: "memory");
#endif
  };

  // ---- prologue: stage first tile -------------------------------------
  {
    const float* bsrc = tileBase(blockIdx.x);   // GEMM_GRID < NTILES always
    if (PHASE1) {
      issueLoads(bsrc);
      storeLds(0);
    } else {
      asyncStage(0, bsrc);
      asyncWait();
    }
  }
  __syncthreads();

  int buf = 0;
  for (int tile = blockIdx.x; tile < NTILES; tile += gridDim.x) {
    const int m0 = tile << 4;
    const int img = m0 / HW_DIM;                // tiles never straddle images
    const int hw0 = m0 - img * HW_DIM;
    const int nextt = tile + (int)gridDim.x;
    const bool hasNext = nextt < NTILES;

    // Kick off next tile's staging before compute (latency overlap).
    if (hasNext) {
      const float* nsrc = tileBase(nextt);
      if (PHASE1) issueLoads(nsrc);             // loads in flight during WMMA
      else        asyncStage(buf ^ 1, nsrc);    // DMA in flight during WMMA
    }

    v8f acc = {0.f, 0.f, 0.f, 0.f, 0.f, 0.f, 0.f, 0.f};
#pragma unroll
    for (int s = 0; s < 64; ++s) {
      v2f B = *(const v2f*)(&sB[buf][(((s << 1) + half) << 5) + (n << 1)]);
      acc = __builtin_amdgcn_wmma_f32_16x16x4_f32(
          /*neg_a=*/false, Areg[s], /*neg_b=*/false, B,
          /*c_mod=*/(short)0, acc, /*reuse_a=*/false, /*reuse_b=*/false);
    }

    // Epilogue.
#pragma unroll
    for (int i = 0; i < 8; ++i) {
      int coI = (w << 4) + i + (half << 3);
      float val = acc[i];
      if (PHASE1) {
        val -= rsub[i];
        val = fminf(fmaxf(val, -cvv[i]), cvv[i]);
        dst[(size_t)coI * M_DIM + m0 + n] = val;
        float vmax = val, vmin = val;
#pragma unroll
        for (int d = 1; d < 16; d <<= 1) {      // reduce across the 16 N-lanes
          vmax = fmaxf(vmax, __shfl_xor(vmax, d, 32));
          vmin = fminf(vmin, __shfl_xor(vmin, d, 32));
        }
        if (n == 0) {
          atomicMax(&smax[coI], fenc(vmax));
          atomicMin(&smin[coI], fenc(vmin));
        }
      } else {
        val += rsub[i];
        dst[((size_t)img * C_DIM + coI) * HW_DIM + hw0 + n] = val;
      }
    }

    // Finish staging the next buffer, then one barrier per tile.
    if (hasNext && PHASE1) storeLds(buf ^ 1);
    if (!PHASE1) asyncWait();
    __syncthreads();
    buf ^= 1;
  }

  if (PHASE1) {
    if (tid < C_DIM) {
      atomicMin(&gmin[tid], smin[tid]);
      atomicMax(&gmax[tid], smax[tid]);
    }
  }
}

// --------------------------------------------------- quantization params
__global__ void k_qparam(const int* gmin, const int* gmax, const int* bwp,
                         float* rowscale, float* rowinv, float* rowdmin,
                         int* rowsame) {
  int c = threadIdx.x;
  int bw = bwp[0];
  float dmax = fdec(gmax[c]);
  float dmin = fdec(gmin[c]);
  int same = (dmax == dmin) || (bw >= 17);
  float rng = same ? 1.f : (dmax - dmin);
  float levels = (bw < 17) ? (float)((1 << bw) - 1) : 1.f;
  rowscale[c] = levels / rng;
  rowinv[c]   = rng / levels;
  rowdmin[c]  = dmin;
  rowsame[c]  = same;
}

// -------------------------------------- quantize P in place + row sums q
__global__ __launch_bounds__(256) void k_quant(float* __restrict__ P,
                                               const float* rowscale,
                                               const float* rowinv,
                                               const float* rowdmin,
                                               const int* rowsame,
                                               float* __restrict__ qsum) {
  __shared__ float red[256];
  int row = blockIdx.x;
  float scale = rowscale[row], inv = rowinv[row], dmin = rowdmin[row];
  int same = rowsame[row];
  float* pr = P + (size_t)row * M_DIM;
  float s = 0.f;
  for (int m = threadIdx.x; m < M_DIM; m += 256) {
    float p = pr[m];
    float q = same ? p : (rintf((p - dmin) * scale) * inv + dmin);  // RNE
    pr[m] = q;
    s += q;
  }
  red[threadIdx.x] = s;
  __syncthreads();
  for (int d = 128; d > 0; d >>= 1) {
    if (threadIdx.x < d) red[threadIdx.x] += red[threadIdx.x + d];
    __syncthreads();
  }
  if (threadIdx.x == 0) qsum[row] = red[0];
}

// ------------------------------------------- off = mn - (u * qsum) / M
__global__ void k_off(const float* __restrict__ u, const float* mn,
                      const float* qsum, float* off) {
  int c = threadIdx.x;
  float s = 0.f;
  for (int k = 0; k < C_DIM; ++k) s += u[c * C_DIM + k] * qsum[k];
  off[c] = mn[c] - s * (1.f / (float)M_DIM);
}

// ---------------------------------------------------------------- launch
extern "C" void kernel_launch(void* const* d_in, const int* in_sizes, int n_in,
                              void* d_out, int out_size, void* d_ws,
                              size_t ws_size, hipStream_t stream) {
  const float* x  = (const float*)d_in[0];
  const float* u  = (const float*)d_in[1];
  const float* cv = (const float*)d_in[2];
  const int*   bw = (const int*)d_in[3];
  float* out = (float*)d_out;

  char* ws = (char*)d_ws;
  float* P = (float*)ws;  // 256 x 100352 f32 = ~103 MB
  size_t small = (size_t)C_DIM * M_DIM * sizeof(float);
  float* sums     = (float*)(ws + small);
  float* mn       = sums + C_DIM;
  float* tvec     = mn + C_DIM;
  float* qsum     = tvec + C_DIM;
  float* offv     = qsum + C_DIM;
  float* rowscale = offv + C_DIM;
  float* rowinv   = rowscale + C_DIM;
  float* rowdmin  = rowinv + C_DIM;
  int*   rowsame  = (int*)(rowdmin + C_DIM);
  int*   gmin     = rowsame + C_DIM;
  int*   gmax     = gmin + C_DIM;

  k_init<<<1, 256, 0, stream>>>(sums, gmin, gmax);
  k_chansum<<<N_IMG * C_DIM, 256, 0, stream>>>(x, sums);
  k_mn_tvec<<<1, 256, 0, stream>>>(u, sums, mn, tvec);
  k_gemm<true><<<GEMM_GRID, 512, 0, stream>>>(x, u, tvec, cv, P, gmin, gmax);
  k_qparam<<<1, 256, 0, stream>>>(gmin, gmax, bw, rowscale, rowinv, rowdmin,
                                  rowsame);
  k_quant<<<256, 256, 0, stream>>>(P, rowscale, rowinv, rowdmin, rowsame, qsum);
  k_off<<<1, 256, 0, stream>>>(u, mn, qsum, offv);
  k_gemm<false><<<GEMM_GRID, 512, 0, stream>>>(P, u, offv, nullptr, out,
                                               nullptr, nullptr);
}